// ShiftAugmentation_18743237280290
// MI455X (gfx1250) — compile-verified
//
#include <hip/hip_runtime.h>
#include <hip/hip_bf16.h>
#include <stdint.h>

// Per-row circular shift (roll): out[i][j] = x[i][(j - s_i) mod T].
// Pure bandwidth-bound data movement: 200 MB @ 23.3 TB/s ~= 8.6 us floor.
// Streamed through LDS with CDNA5 async global<->LDS DMA ops (ASYNCcnt),
// data never touches VGPRs. Fast path exploits that the async GVS form adds
// INST_OFFSET to BOTH the LDS and global addresses, so a whole 16 KB tile
// needs only one base-address computation per side.

#define ROLL_BLOCK 256
#define ROLL_ITERS 16                      // b32 async loads per thread
#define ROLL_TILE  (ROLL_BLOCK * ROLL_ITERS) // 4096 dwords / 16 KB per tile

// Fast-path async ops with literal instruction offsets (offset applies to
// both LDS and global sides equally; both advance by the same amount here).
#define ROLL_LD(OFS) \
    asm volatile("global_load_async_to_lds_b32 %0, %1, %2 offset:" OFS \
                 :: "v"(la), "v"(voff), "s"(xrow) : "memory")
#define ROLL_ST(OFS) \
    asm volatile("global_store_async_from_lds_b128 %0, %1, %2 offset:" OFS \
                 :: "v"(voff2), "v"(la2), "s"(orow) : "memory")

__global__ __launch_bounds__(ROLL_BLOCK)
void ShiftAugmentation_18743237280290_kernel(const float* __restrict__ x,
                                             const int* __restrict__ shifts,
                                             float* __restrict__ out,
                                             int T)
{
    __shared__ float buf[ROLL_TILE]; // 16 KB of the 320 KB/WGP LDS

    const int row      = blockIdx.y;
    const int tilebase = blockIdx.x * ROLL_TILE;
    const int tid      = threadIdx.x;

    int s = shifts[row] % T;        // one scalar division per block
    if (s < 0) s += T;

    const float* xrow = x   + (size_t)row * (size_t)T; // uniform -> SGPR base
    float*       orow = out + (size_t)row * (size_t)T; // uniform -> SGPR base

    // LDS byte offset of buf: low 32 bits of the flat shared address are the
    // LDS offset on CDNA5 (aperture lives in addr[63:32]).
    const uint32_t lds0 = (uint32_t)(uintptr_t)(&buf[0]);

    // Uniform per-block fast-path test: no wrap (j >= s for all j in tile)
    // and no tail predication (whole tile in-bounds).
    const bool fast = (s <= tilebase) && (tilebase + ROLL_TILE <= T);

    if (fast) {
        // ---- Fast path: 16x async b32 load, 4x async b128 store ----
        // Load: LDS dword (k*256 + tid) <- x[tilebase + k*256 + tid - s].
        // Per-k advance is 1024 B on BOTH sides -> fold into offset imm.
        uint32_t voff = ((uint32_t)(tilebase + tid - s)) << 2;
        uint32_t la   = lds0 + ((uint32_t)tid << 2);
        ROLL_LD("0");     ROLL_LD("1024");  ROLL_LD("2048");  ROLL_LD("3072");
        ROLL_LD("4096");  ROLL_LD("5120");  ROLL_LD("6144");  ROLL_LD("7168");
        ROLL_LD("8192");  ROLL_LD("9216");  ROLL_LD("10240"); ROLL_LD("11264");
        ROLL_LD("12288"); ROLL_LD("13312"); ROLL_LD("14336"); ROLL_LD("15360");

        asm volatile("s_wait_asynccnt 0x0" ::: "memory");

        // Store: lane covers 4 consecutive dwords, 16 B aligned on both the
        // LDS side (tid*16) and the global side (tilebase%4096==0).
        uint32_t voff2 = ((uint32_t)(tilebase + tid * 4)) << 2;
        uint32_t la2   = lds0 + ((uint32_t)tid << 4);
        ROLL_ST("0"); ROLL_ST("4096"); ROLL_ST("8192"); ROLL_ST("12288");
    } else {
        // ---- Slow path (wrap tile / tail tile only: ~2 of 25 per row) ----
#pragma unroll
        for (int k = 0; k < ROLL_ITERS; ++k) {
            int j = tilebase + k * ROLL_BLOCK + tid;
            if (j < T) {
                int src = j - s;
                if (src < 0) src += T;                  // wrap
                uint32_t voff = (uint32_t)src << 2;
                uint32_t la   = lds0 + ((uint32_t)(k * ROLL_BLOCK + tid) << 2);
                asm volatile("global_load_async_to_lds_b32 %0, %1, %2"
                             :: "v"(la), "v"(voff), "s"(xrow) : "memory");
            }
        }
        asm volatile("s_wait_asynccnt 0x0" ::: "memory");
#pragma unroll
        for (int k = 0; k < ROLL_ITERS; ++k) {
            int j = tilebase + k * ROLL_BLOCK + tid;
            if (j < T) {
                uint32_t voff = (uint32_t)j << 2;
                uint32_t la   = lds0 + ((uint32_t)(k * ROLL_BLOCK + tid) << 2);
                asm volatile("global_store_async_from_lds_b32 %0, %1, %2"
                             :: "v"(voff), "v"(la), "s"(orow) : "memory");
            }
        }
    }

    // s_endpgm does an implicit wait-idle, but drain explicitly anyway.
    asm volatile("s_wait_asynccnt 0x0" ::: "memory");
}

extern "C" void kernel_launch(void* const* d_in, const int* in_sizes, int n_in,
                              void* d_out, int out_size, void* d_ws, size_t ws_size,
                              hipStream_t stream)
{
    const float* x      = (const float*)d_in[0];
    const int*   shifts = (const int*)d_in[1];
    float*       out    = (float*)d_out;

    const int B = in_sizes[1];                 // 256 rows (one shift per row)
    const int T = in_sizes[0] / B;             // 100000 elements per row

    const int tiles = (T + ROLL_TILE - 1) / ROLL_TILE;
    dim3 grid(tiles, B);
    ShiftAugmentation_18743237280290_kernel<<<grid, ROLL_BLOCK, 0, stream>>>(
        x, shifts, out, T);
}